// MQ_2662879724284
// MI455X (gfx1250) — compile-verified
//
#include <hip/hip_runtime.h>

// ---------------------------------------------------------------------------
// MI455X / gfx1250 VQ-VAE forward pass, fp32 WMMA (V_WMMA_F32_16X16X4_F32),
// LDS-staged tiled GEMMs (64x64 macro-tile per 4-wave block).
//
// Input index map (setup_inputs() dict order):
//  0 x(8192x512) 1 query_tokens(4x512) 2 codebooks(4x4096x64)
//  3 enc_W1(4x512x128) 4 enc_b1(4x128) 5 enc_W2(4x128x256) 6 enc_b2(4x256)
//  7 enc_W3(4x256x64)  8 enc_b3(4x64)
//  9 enc_g1(4x128) 10 enc_beta1(4x128) 11 enc_g2(4x256) 12 enc_beta2(4x256)
// 13 dec_W1(64x256) 14 dec_b1(256) 15 dec_W2(256x128) 16 dec_b2(128)
// 17 dec_W3(128x512) 18 dec_b3(512)
// 19 dec_g1(256) 20 dec_beta1(256) 21 dec_g2(128) 22 dec_beta2(128)
//
// d_out: [x_hat 8192*512][ze 4*8192*64][ce 4*8192*64]  (float32)
// ---------------------------------------------------------------------------

typedef __attribute__((ext_vector_type(2))) float v2f;
typedef __attribute__((ext_vector_type(8))) float v8f;

#define MBOOK   4
#define IN_DIM  512
#define ZDIM    64
#define NEMB    4096
#define BATCH   8192
#define BN_EPS  1e-5f
#define TILE_K  16
#define BPAD    68   // row pitch of LDS B tile: keeps ds_store_b128 16B-aligned
                     // and makes compute reads bank-conflict-free (4*l16+k)

// ---------------------------------------------------------------------------
// WMMA f32 16x16x4 operand layout (wave32):
//   half = lane>>4 selects K pair {0,1} vs {2,3}; l16 = lane&15 is A-row/B-col.
//   D: VGPR r -> row (r + 8*half), col l16.
// ---------------------------------------------------------------------------

// Tiled GEMM: C[M,N] = A[M,K] @ B[K,N] + bias[N], batched over blockIdx.z with
// element strides. Block = 128 threads (4 waves) -> 64x64 C macro-tile:
// wave w owns rows [w*16, w*16+16), all 64 cols (4 v8f accumulators).
// B panel (16 x 64) is staged in LDS per K-chunk and shared by the 4 waves;
// the per-k A fragment (global_load_b64) is reused across the 4 column tiles.
// MASKED variant implements encoder layer 1's virtual input:
//   a[e] = x[row][e] * (e/128 == book) + qt[book][e]
template<bool MASKED>
__global__ __launch_bounds__(128)
void gemm_tile(const float* __restrict__ A, long long sA,
               const float* __restrict__ B, long long sB,
               const float* __restrict__ bias, long long sBias,
               float* __restrict__ C, long long sC,
               const float* __restrict__ qt,
               int M, int N, int K)
{
    const int g = blockIdx.z;
    A    += (size_t)g * sA;          // MASKED: sA==0, A is the shared x
    B    += (size_t)g * sB;
    bias += (size_t)g * sBias;
    C    += (size_t)g * sC;
    const float* qm = MASKED ? (qt + (size_t)g * K) : nullptr;

    const int lane = threadIdx.x & 31;
    const int wave = threadIdx.x >> 5;
    const int half = lane >> 4;
    const int l16  = lane & 15;

    const int mbase = blockIdx.x * 64 + wave * 16;
    const int nbase = blockIdx.y * 64;
    const int arow  = mbase + l16;

    __shared__ float Bt[TILE_K][BPAD];

    v8f acc[4] = {};

    const int ldr = threadIdx.x >> 4;         // 0..7  (staging row)
    const int ldc = (threadIdx.x & 15) * 4;   // 0..60 (staging col, float4)

    for (int k0 = 0; k0 < K; k0 += TILE_K) {
        __syncthreads();                      // previous chunk's reads done
#pragma unroll
        for (int p = 0; p < 2; ++p) {
            const int rr = ldr + p * 8;
            *(float4*)&Bt[rr][ldc] =
                *(const float4*)&B[(size_t)(k0 + rr) * N + nbase + ldc];
        }
        __syncthreads();

#pragma unroll
        for (int ks = 0; ks < TILE_K; ks += 4) {
            const int klo = k0 + ks + 2 * half;
            const int kt  = ks + 2 * half;
            v2f a;
            if (MASKED) {
                const float m0 = ((klo       >> 7) == g) ? 1.0f : 0.0f;
                const float m1 = (((klo + 1) >> 7) == g) ? 1.0f : 0.0f;
                a.x = A[(size_t)arow * K + klo]     * m0 + qm[klo];
                a.y = A[(size_t)arow * K + klo + 1] * m1 + qm[klo + 1];
            } else {
                a.x = A[(size_t)arow * K + klo];
                a.y = A[(size_t)arow * K + klo + 1];
            }
#pragma unroll
            for (int ct = 0; ct < 4; ++ct) {
                v2f b;
                b.x = Bt[kt][ct * 16 + l16];
                b.y = Bt[kt + 1][ct * 16 + l16];
                acc[ct] = __builtin_amdgcn_wmma_f32_16x16x4_f32(
                    false, a, false, b, (short)0, acc[ct], false, false);
            }
        }
    }

#pragma unroll
    for (int ct = 0; ct < 4; ++ct) {
        const int col = nbase + ct * 16 + l16;
        const float bv = bias[col];
#pragma unroll
        for (int r = 0; r < 8; ++r)
            C[(size_t)(mbase + r + half * 8) * N + col] = acc[ct][r] + bv;
    }
}

// BN statistics over the batch axis. h layout [G][R][F]; one block per (g,f):
// scale = gamma * rsqrt(var+eps); shift = beta - mu*scale.
__global__ __launch_bounds__(256)
void bn_stats(const float* __restrict__ h, const float* __restrict__ gamma,
              const float* __restrict__ beta, float* __restrict__ scale,
              float* __restrict__ shift, int R, int F)
{
    const int f = blockIdx.x;
    const int g = blockIdx.y;
    const float* p = h + (size_t)g * R * F + f;

    float s = 0.f, s2 = 0.f;
    for (int n = threadIdx.x; n < R; n += blockDim.x) {
        const float v = p[(size_t)n * F];
        s += v; s2 += v * v;
    }
    __shared__ float sh[256], sh2[256];
    sh[threadIdx.x] = s; sh2[threadIdx.x] = s2;
    __syncthreads();
    for (int st = blockDim.x >> 1; st > 0; st >>= 1) {
        if ((int)threadIdx.x < st) {
            sh[threadIdx.x]  += sh[threadIdx.x + st];
            sh2[threadIdx.x] += sh2[threadIdx.x + st];
        }
        __syncthreads();
    }
    if (threadIdx.x == 0) {
        const float mu  = sh[0] / (float)R;
        const float var = sh2[0] / (float)R - mu * mu;
        const float sc  = gamma[(size_t)g * F + f] * rsqrtf(var + BN_EPS);
        scale[(size_t)g * F + f] = sc;
        shift[(size_t)g * F + f] = beta[(size_t)g * F + f] - mu * sc;
    }
}

// In-place BN apply + ReLU over [G][R][F] flat; RF = R*F.
__global__ __launch_bounds__(256)
void bn_apply_relu(float* __restrict__ h, const float* __restrict__ scale,
                   const float* __restrict__ shift, int total, int F, int RF)
{
    const int t = blockIdx.x * blockDim.x + threadIdx.x;
    if (t >= total) return;
    const int f = t % F;
    const int g = t / RF;
    const float v = h[t] * scale[g * F + f] + shift[g * F + f];
    h[t] = v > 0.f ? v : 0.f;
}

// Half squared code norms: cnorm[m*4096+j] = 0.5 * |codebooks[m][j]|^2
__global__ __launch_bounds__(256)
void cnorm_kernel(const float* __restrict__ cb, float* __restrict__ cnorm)
{
    const int t = blockIdx.x * blockDim.x + threadIdx.x;
    if (t >= MBOOK * NEMB) return;
    const float* c = cb + (size_t)t * ZDIM;
    float s = 0.f;
#pragma unroll
    for (int k = 0; k < ZDIM; ++k) s += c[k] * c[k];
    cnorm[t] = 0.5f * s;
}

// Fused VQ nearest-code search: argmin_j |ze-c_j|^2 == argmax_j ze.c_j-0.5|c_j|^2.
// Block = 4 waves / 64 batch rows. Per 16-code tile: codebook rows staged in
// LDS once (shared by the 4 waves, ds_load_b64 on the compute side), ze tile
// preloaded in registers, 16 WMMAs, fused running-argmax, then a width-16
// shuffle reduction across the lanes sharing each output row. wave32.
__global__ __launch_bounds__(128)
void vq_kernel(const float* __restrict__ ze, const float* __restrict__ cb,
               const float* __restrict__ cnorm, int* __restrict__ idxout)
{
    const int m    = blockIdx.z;
    const int lane = threadIdx.x & 31;
    const int wave = threadIdx.x >> 5;
    const int half = lane >> 4;
    const int l16  = lane & 15;
    const int row0 = blockIdx.x * 64 + wave * 16;

    const float* zem = ze + (size_t)m * BATCH * ZDIM;
    const float* cbm = cb + (size_t)m * NEMB * ZDIM;
    const float* cnm = cnorm + (size_t)m * NEMB;

    __shared__ float Ct[16][BPAD];           // [code][k], padded

    // Preload this wave's A tile (16 rows x K=64) once: 16 k-steps.
    v2f a[16];
    const int arow = row0 + l16;
#pragma unroll
    for (int kk = 0; kk < 16; ++kk) {
        const int klo = kk * 4 + 2 * half;
        a[kk].x = zem[(size_t)arow * ZDIM + klo];
        a[kk].y = zem[(size_t)arow * ZDIM + klo + 1];
    }

    float bs[8]; int bi[8];
#pragma unroll
    for (int r = 0; r < 8; ++r) { bs[r] = -3.402823466e38f; bi[r] = 0; }

    const int scode = threadIdx.x >> 4;        // 0..7 staging code row
    const int sc4   = (threadIdx.x & 15) * 4;  // staging k (float4)

    for (int j0 = 0; j0 < NEMB; j0 += 16) {
        __syncthreads();
#pragma unroll
        for (int p = 0; p < 2; ++p) {
            const int code = scode + p * 8;
            *(float4*)&Ct[code][sc4] =
                *(const float4*)&cbm[(size_t)(j0 + code) * ZDIM + sc4];
            if (j0 + 16 < NEMB)   // prefetch next tile (global_prefetch path)
                __builtin_prefetch(&cbm[(size_t)(j0 + 16 + code) * ZDIM + sc4], 0, 1);
        }
        __syncthreads();

        v8f acc = {};
#pragma unroll
        for (int kk = 0; kk < 16; ++kk) {
            const int klo = kk * 4 + 2 * half;
            v2f b;
            b.x = Ct[l16][klo];                // adjacent -> ds_load_b64
            b.y = Ct[l16][klo + 1];
            acc = __builtin_amdgcn_wmma_f32_16x16x4_f32(
                false, a[kk], false, b, (short)0, acc, false, false);
        }
        const int jcol = j0 + l16;
        const float cn = cnm[jcol];
#pragma unroll
        for (int r = 0; r < 8; ++r) {
            const float sc = acc[r] - cn;
            if (sc > bs[r]) { bs[r] = sc; bi[r] = jcol; }
        }
    }

    // Reduce across the 16 lanes that share each output row.
#pragma unroll
    for (int r = 0; r < 8; ++r) {
        float s = bs[r]; int i = bi[r];
        for (int off = 8; off >= 1; off >>= 1) {
            const float os = __shfl_xor(s, off, 16);
            const int   oi = __shfl_xor(i, off, 16);
            if (os > s || (os == s && oi < i)) { s = os; i = oi; }
        }
        if (l16 == 0)
            idxout[(size_t)m * BATCH + row0 + r + half * 8] = i;
    }
}

// ce[m][n][:] = codebooks[m][idx[m][n]][:];  zq[n][:] = sum_m ce[m][n][:]
__global__ __launch_bounds__(256)
void gather_zq(const float* __restrict__ cb, const int* __restrict__ idx,
               float* __restrict__ ce, float* __restrict__ zq)
{
    const int t = blockIdx.x * blockDim.x + threadIdx.x;   // over BATCH*ZDIM
    if (t >= BATCH * ZDIM) return;
    const int n = t >> 6;
    const int k = t & 63;
    float s = 0.f;
#pragma unroll
    for (int m = 0; m < MBOOK; ++m) {
        const int j = idx[(size_t)m * BATCH + n];
        const float v = cb[((size_t)m * NEMB + j) * ZDIM + k];
        ce[((size_t)m * BATCH + n) * ZDIM + k] = v;
        s += v;
    }
    zq[t] = s;
}

// ---------------------------------------------------------------------------

extern "C" void kernel_launch(void* const* d_in, const int* in_sizes, int n_in,
                              void* d_out, int out_size, void* d_ws, size_t ws_size,
                              hipStream_t stream)
{
    const float* x         = (const float*)d_in[0];
    const float* qt        = (const float*)d_in[1];
    const float* codebooks = (const float*)d_in[2];
    const float* enc_W1 = (const float*)d_in[3];
    const float* enc_b1 = (const float*)d_in[4];
    const float* enc_W2 = (const float*)d_in[5];
    const float* enc_b2 = (const float*)d_in[6];
    const float* enc_W3 = (const float*)d_in[7];
    const float* enc_b3 = (const float*)d_in[8];
    const float* enc_g1 = (const float*)d_in[9];
    const float* enc_be1= (const float*)d_in[10];
    const float* enc_g2 = (const float*)d_in[11];
    const float* enc_be2= (const float*)d_in[12];
    const float* dec_W1 = (const float*)d_in[13];
    const float* dec_b1 = (const float*)d_in[14];
    const float* dec_W2 = (const float*)d_in[15];
    const float* dec_b2 = (const float*)d_in[16];
    const float* dec_W3 = (const float*)d_in[17];
    const float* dec_b3 = (const float*)d_in[18];
    const float* dec_g1 = (const float*)d_in[19];
    const float* dec_be1= (const float*)d_in[20];
    const float* dec_g2 = (const float*)d_in[21];
    const float* dec_be2= (const float*)d_in[22];
    (void)in_sizes; (void)n_in; (void)out_size; (void)ws_size;

    // Output regions
    float* xhat = (float*)d_out;                       // 8192*512
    float* ze   = xhat + (size_t)BATCH * IN_DIM;       // 4*8192*64
    float* ce   = ze + (size_t)MBOOK * BATCH * ZDIM;   // 4*8192*64

    // Workspace carve-out (floats)
    float* ws = (float*)d_ws;
    size_t o = 0;
    float* h1    = ws + o; o += (size_t)MBOOK * BATCH * 128;
    float* h2    = ws + o; o += (size_t)MBOOK * BATCH * 256;
    float* zq    = ws + o; o += (size_t)BATCH * ZDIM;
    float* dh1   = ws + o; o += (size_t)BATCH * 256;
    float* dh2   = ws + o; o += (size_t)BATCH * 128;
    float* cnrm  = ws + o; o += (size_t)MBOOK * NEMB;
    float* scale = ws + o; o += 1024;
    float* shift = ws + o; o += 1024;
    int*   idxb  = (int*)(ws + o);                     // 4*8192 ints

    const int GX = BATCH / 64;   // 128 row-tiles

    // ---- Encoder layer 1: (4) 8192x128 = xm @ W1, K=512 ----
    gemm_tile<true><<<dim3(GX, 128 / 64, MBOOK), 128, 0, stream>>>(
        x, 0, enc_W1, 512LL * 128, enc_b1, 128, h1, (long long)BATCH * 128,
        qt, BATCH, 128, IN_DIM);
    bn_stats<<<dim3(128, MBOOK), 256, 0, stream>>>(h1, enc_g1, enc_be1,
                                                   scale, shift, BATCH, 128);
    bn_apply_relu<<<(MBOOK * BATCH * 128) / 256, 256, 0, stream>>>(
        h1, scale, shift, MBOOK * BATCH * 128, 128, BATCH * 128);

    // ---- Encoder layer 2: (4) 8192x256 = h1 @ W2, K=128 ----
    gemm_tile<false><<<dim3(GX, 256 / 64, MBOOK), 128, 0, stream>>>(
        h1, (long long)BATCH * 128, enc_W2, 128LL * 256, enc_b2, 256,
        h2, (long long)BATCH * 256, nullptr, BATCH, 256, 128);
    bn_stats<<<dim3(256, MBOOK), 256, 0, stream>>>(h2, enc_g2, enc_be2,
                                                   scale, shift, BATCH, 256);
    bn_apply_relu<<<(MBOOK * BATCH * 256) / 256, 256, 0, stream>>>(
        h2, scale, shift, MBOOK * BATCH * 256, 256, BATCH * 256);

    // ---- Encoder layer 3: (4) 8192x64 = h2 @ W3, K=256 -> ze (output) ----
    gemm_tile<false><<<dim3(GX, ZDIM / 64, MBOOK), 128, 0, stream>>>(
        h2, (long long)BATCH * 256, enc_W3, 256LL * ZDIM, enc_b3, ZDIM,
        ze, (long long)BATCH * ZDIM, nullptr, BATCH, ZDIM, 256);

    // ---- VQ: code norms, fused WMMA distance + argmin, gather + zq ----
    cnorm_kernel<<<(MBOOK * NEMB) / 256, 256, 0, stream>>>(codebooks, cnrm);
    vq_kernel<<<dim3(GX, 1, MBOOK), 128, 0, stream>>>(ze, codebooks, cnrm, idxb);
    gather_zq<<<(BATCH * ZDIM) / 256, 256, 0, stream>>>(codebooks, idxb, ce, zq);

    // ---- Decoder layer 1: 8192x256 = zq @ dec_W1, K=64 ----
    gemm_tile<false><<<dim3(GX, 256 / 64, 1), 128, 0, stream>>>(
        zq, 0, dec_W1, 0, dec_b1, 0, dh1, 0, nullptr, BATCH, 256, ZDIM);
    bn_stats<<<dim3(256, 1), 256, 0, stream>>>(dh1, dec_g1, dec_be1,
                                               scale, shift, BATCH, 256);
    bn_apply_relu<<<(BATCH * 256) / 256, 256, 0, stream>>>(
        dh1, scale, shift, BATCH * 256, 256, BATCH * 256);

    // ---- Decoder layer 2: 8192x128 = dh1 @ dec_W2, K=256 ----
    gemm_tile<false><<<dim3(GX, 128 / 64, 1), 128, 0, stream>>>(
        dh1, 0, dec_W2, 0, dec_b2, 0, dh2, 0, nullptr, BATCH, 128, 256);
    bn_stats<<<dim3(128, 1), 256, 0, stream>>>(dh2, dec_g2, dec_be2,
                                               scale, shift, BATCH, 128);
    bn_apply_relu<<<(BATCH * 128) / 256, 256, 0, stream>>>(
        dh2, scale, shift, BATCH * 128, 128, BATCH * 128);

    // ---- Decoder layer 3: 8192x512 = dh2 @ dec_W3, K=128 -> x_hat ----
    gemm_tile<false><<<dim3(GX, IN_DIM / 64, 1), 128, 0, stream>>>(
        dh2, 0, dec_W3, 0, dec_b3, 0, xhat, 0, nullptr, BATCH, IN_DIM, 128);
}